// LinearAttention_59012850647694
// MI455X (gfx1250) — compile-verified
//
#include <hip/hip_runtime.h>
#include <hip/hip_bf16.h>
#include <cstdint>

// ---------------------------------------------------------------------------
// LinearAttention for MI455X (gfx1250), fp32 end-to-end via V_WMMA_F32_16X16X4_F32
// B=4, S=4096, HID=1024, H=8, D=128, CHUNK=64
// ---------------------------------------------------------------------------

#define HIDDEN 1024
#define SEQ    4096
#define NB     4
#define NH     8
#define DH     128
#define CH     64
#define ROWS   (NB * SEQ)        // 16384

typedef __attribute__((ext_vector_type(2))) float v2f;
typedef __attribute__((ext_vector_type(8))) float v8f;
typedef __attribute__((ext_vector_type(4))) int   v4i;

#define AS1 __attribute__((address_space(1)))
#define AS3 __attribute__((address_space(3)))

// CDNA5 async memory->LDS path (ASYNCcnt-tracked), with sync fallback if the
// toolchain doesn't declare the builtins. Probe round 2 confirmed the builtin
// exists with signature (v4i AS1* src, <lds ptr> dst, int offset, int cpol).
#if defined(__gfx1250__) &&                                            \
    __has_builtin(__builtin_amdgcn_global_load_async_to_lds_b128) &&   \
    __has_builtin(__builtin_amdgcn_s_wait_asynccnt)
#define USE_ASYNC_LDS 1
#else
#define USE_ASYNC_LDS 0
#endif

__device__ __forceinline__ v8f wmma4(v2f a, v2f b, v8f c) {
  // D = A(16x4) * B(4x16) + C(16x16), all fp32
  return __builtin_amdgcn_wmma_f32_16x16x4_f32(
      /*neg_a=*/false, a, /*neg_b=*/false, b,
      /*c_mod=*/(short)0, c, /*reuse_a=*/false, /*reuse_b=*/false);
}

// ---------------------------------------------------------------------------
// Y[m,n] = sum_k X[m,k] * W[n,k]     (X: [M,1024], W: [1024,1024] row-major)
// grid: (M/64, 16), block: 128 (4 waves). Wave w -> m-tile blockIdx.x*4+w,
// n-tiles blockIdx.y*4 .. +3 (64 columns).
// ---------------------------------------------------------------------------
__global__ __launch_bounds__(128) void gemm_xwt_kernel(
    const float* __restrict__ X, const float* __restrict__ W,
    float* __restrict__ Y) {
  const int lane = threadIdx.x & 31;
  const int wave = threadIdx.x >> 5;
  const int row  = lane & 15;       // M (for A) / N (for B/C) index within tile
  const int hi   = lane >> 4;       // half-wave selector -> k offset 2*hi

  const int m0  = (blockIdx.x * 4 + wave) * 16;
  const int nt0 = blockIdx.y * 4;   // first of four 16-wide n-tiles

  const float* xa = X + (size_t)(m0 + row) * HIDDEN + 2 * hi;
  const float* wb = W + (size_t)(nt0 * 16 + row) * HIDDEN + 2 * hi;

  v8f acc[4];
  acc[0] = {}; acc[1] = {}; acc[2] = {}; acc[3] = {};

#pragma unroll 4
  for (int k = 0; k < HIDDEN; k += 4) {
    v2f a = *(const v2f*)(xa + k);
    v2f b0 = *(const v2f*)(wb + k);
    v2f b1 = *(const v2f*)(wb + 16 * HIDDEN + k);
    v2f b2 = *(const v2f*)(wb + 32 * HIDDEN + k);
    v2f b3 = *(const v2f*)(wb + 48 * HIDDEN + k);
    acc[0] = wmma4(a, b0, acc[0]);
    acc[1] = wmma4(a, b1, acc[1]);
    acc[2] = wmma4(a, b2, acc[2]);
    acc[3] = wmma4(a, b3, acc[3]);
  }

  // C/D layout: vgpr j, lanes0-15 -> M=j, lanes16-31 -> M=j+8; N = lane%16
  const size_t ybase = (size_t)(m0 + 8 * hi) * HIDDEN + (size_t)nt0 * 16 + row;
#pragma unroll
  for (int t = 0; t < 4; ++t)
#pragma unroll
    for (int j = 0; j < 8; ++j)
      Y[ybase + (size_t)j * HIDDEN + t * 16] = acc[t][j];
}

// ---------------------------------------------------------------------------
// Chunked causal linear attention + denom + rmsnorm, one workgroup per (b,h).
// 256 threads = 8 waves. State kept transposed in LDS: sst[n=dcol][k=drow].
// LDS strides padded (132 / 68) to avoid bank conflicts.
// ---------------------------------------------------------------------------
#define QSTRIDE 132
#define SSTRIDE 68
#define SMEM_FLOATS (3 * CH * QSTRIDE + DH * QSTRIDE + CH * SSTRIDE + DH + CH + CH)

__global__ __launch_bounds__(256) void linattn_kernel(
    const float* __restrict__ Q, const float* __restrict__ K,
    const float* __restrict__ V, const float* __restrict__ normw,
    float* __restrict__ O) {
  extern __shared__ float smem[];
  float* sq  = smem;                       // [64][132]  (reused as o-buffer)
  float* sk  = sq  + CH * QSTRIDE;         // [64][132]
  float* sv  = sk  + CH * QSTRIDE;         // [64][132]
  float* sst = sv  + CH * QSTRIDE;         // [128][132] transposed state
  float* ssc = sst + DH * QSTRIDE;         // [64][68]   masked scores
  float* zst = ssc + CH * SSTRIDE;         // [128] running column-sum of K
  float* sdn = zst + DH;                   // [64] denominators
  float* snr = sdn + CH;                   // [64] rms scales

  const int bh = blockIdx.x;
  const int b = bh >> 3, h = bh & 7;
  const size_t base = (size_t)b * SEQ * HIDDEN + (size_t)h * DH;

  const int tid  = threadIdx.x;
  const int lane = tid & 31;
  const int wave = tid >> 5;
  const int row  = lane & 15;
  const int hi   = lane >> 4;

  const int mt    = wave & 3;        // 16-row tile of the 64-row chunk
  const int nh4   = wave >> 2;       // 0/1 -> which 64-col half of D
  const int nbase = nh4 * 64;

  // zero state + z
  for (int i = tid; i < DH * QSTRIDE; i += 256) sst[i] = 0.f;
  if (tid < DH) zst[tid] = 0.f;
  __syncthreads();

  for (int chunk = 0; chunk < SEQ / CH; ++chunk) {
    const int s0 = chunk * CH;

    // ---- stage Q,K,V chunk tiles (64x128 each) into LDS ----
#if USE_ASYNC_LDS
    for (int i = tid; i < CH * DH / 4; i += 256) {
      const int r  = i >> 5;
      const int c4 = (i & 31) * 4;
      const size_t g = base + (size_t)(s0 + r) * HIDDEN + c4;
      __builtin_amdgcn_global_load_async_to_lds_b128(
          (AS1 v4i*)(Q + g), (AS3 v4i*)(sq + r * QSTRIDE + c4), 0, 0);
      __builtin_amdgcn_global_load_async_to_lds_b128(
          (AS1 v4i*)(K + g), (AS3 v4i*)(sk + r * QSTRIDE + c4), 0, 0);
      __builtin_amdgcn_global_load_async_to_lds_b128(
          (AS1 v4i*)(V + g), (AS3 v4i*)(sv + r * QSTRIDE + c4), 0, 0);
    }
    __builtin_amdgcn_s_wait_asynccnt(0);
#else
    for (int i = tid; i < CH * DH / 4; i += 256) {
      const int r  = i >> 5;
      const int c4 = (i & 31) * 4;
      const size_t g = base + (size_t)(s0 + r) * HIDDEN + c4;
      *(float4*)(sq + r * QSTRIDE + c4) = *(const float4*)(Q + g);
      *(float4*)(sk + r * QSTRIDE + c4) = *(const float4*)(K + g);
      *(float4*)(sv + r * QSTRIDE + c4) = *(const float4*)(V + g);
    }
#endif
    __syncthreads();

    // ---- masked scores = tril(Q @ K^T)  (wave owns tiles (mt, ct0..ct0+1)) ----
    {
      const int ct0 = nh4 * 2;
#pragma unroll
      for (int t = 0; t < 2; ++t) {
        const int ct = ct0 + t;
        v8f sc = {};
        if (ct <= mt) {
          const float* ap = sq + (mt * 16 + row) * QSTRIDE + 2 * hi;
          const float* bp = sk + (ct * 16 + row) * QSTRIDE + 2 * hi;
          for (int dk = 0; dk < DH; dk += 4) {
            v2f a  = *(const v2f*)(ap + dk);
            v2f bb = *(const v2f*)(bp + dk);
            sc = wmma4(a, bb, sc);
          }
          if (ct == mt) {
#pragma unroll
            for (int j = 0; j < 8; ++j)
              if (row > j + 8 * hi) sc[j] = 0.f;
          }
        }
#pragma unroll
        for (int j = 0; j < 8; ++j)
          ssc[(mt * 16 + j + 8 * hi) * SSTRIDE + ct * 16 + row] = sc[j];
      }
    }
    __syncthreads();

    // ---- o = Q @ state + scores @ V   (wave owns 4 tiles: rows mt, cols nbase..) ----
    v8f acc[4];
    acc[0] = {}; acc[1] = {}; acc[2] = {}; acc[3] = {};
    {
      const float* ap = sq + (mt * 16 + row) * QSTRIDE + 2 * hi;
      for (int dk = 0; dk < DH; dk += 4) {
        v2f a = *(const v2f*)(ap + dk);
#pragma unroll
        for (int j = 0; j < 4; ++j) {
          v2f bb = *(const v2f*)(sst + (nbase + j * 16 + row) * QSTRIDE + dk + 2 * hi);
          acc[j] = wmma4(a, bb, acc[j]);
        }
      }
      const float* ap2 = ssc + (mt * 16 + row) * SSTRIDE + 2 * hi;
      for (int c = 0; c < CH; c += 4) {
        v2f a = *(const v2f*)(ap2 + c);
        const int r0 = c + 2 * hi;
#pragma unroll
        for (int j = 0; j < 4; ++j) {
          v2f bb;
          bb[0] = sv[r0 * QSTRIDE + nbase + j * 16 + row];
          bb[1] = sv[(r0 + 1) * QSTRIDE + nbase + j * 16 + row];
          acc[j] = wmma4(a, bb, acc[j]);
        }
      }
    }

    // ---- denominator: |Q.z_start + rowsum(masked scores)| clamped at 1 ----
    if (tid < CH) {
      float qz = 0.f;
      for (int d = 0; d < DH; ++d) qz += sq[tid * QSTRIDE + d] * zst[d];
      float rs = 0.f;
      for (int t = 0; t < CH; ++t) rs += ssc[tid * SSTRIDE + t];
      float dn = fabsf(qz + rs);
      sdn[tid] = dn > 1.0f ? dn : 1.0f;
    }
    __syncthreads();  // sdn ready; all reads of sst/sq for this chunk done

    // ---- o/denom into sq (reuse as o-buffer) ----
#pragma unroll
    for (int t = 0; t < 4; ++t)
#pragma unroll
      for (int j = 0; j < 8; ++j) {
        const int rl = mt * 16 + j + 8 * hi;
        sq[rl * QSTRIDE + nbase + t * 16 + row] = acc[t][j] / sdn[rl];
      }

    // ---- z_start += column sums of K chunk ----
    if (tid < DH) {
      float zs = 0.f;
      for (int c = 0; c < CH; ++c) zs += sk[c * QSTRIDE + tid];
      zst[tid] += zs;
    }

    // ---- state += K^T @ V   (wave owns m-tile = wave, all 8 n-tiles) ----
    {
      v8f dacc[8];
#pragma unroll
      for (int j = 0; j < 8; ++j) dacc[j] = {};
      const int dm0 = wave * 16;
      for (int c = 0; c < CH; c += 4) {
        const int r0 = c + 2 * hi;
        v2f a;
        a[0] = sk[r0 * QSTRIDE + dm0 + row];
        a[1] = sk[(r0 + 1) * QSTRIDE + dm0 + row];
#pragma unroll
        for (int j = 0; j < 8; ++j) {
          v2f bb;
          bb[0] = sv[r0 * QSTRIDE + j * 16 + row];
          bb[1] = sv[(r0 + 1) * QSTRIDE + j * 16 + row];
          dacc[j] = wmma4(a, bb, dacc[j]);
        }
      }
      // accumulate transposed: element (m=dm0+jj+8*hi, n=j*16+row) -> sst[n][m]
#pragma unroll
      for (int j = 0; j < 8; ++j) {
        float* p = sst + (j * 16 + row) * QSTRIDE + dm0 + 8 * hi;
#pragma unroll
        for (int jj = 0; jj < 8; ++jj) p[jj] += dacc[j][jj];
      }
    }
    __syncthreads();  // o-buffer + state update complete

    // ---- rmsnorm scale per row ----
    if (tid < CH) {
      float ss = 0.f;
      for (int d = 0; d < DH; ++d) {
        const float x = sq[tid * QSTRIDE + d];
        ss += x * x;
      }
      snr[tid] = rsqrtf(ss * (1.0f / DH) + 1e-5f);
    }
    __syncthreads();

    // ---- write normalized o to global ----
    for (int i = tid; i < CH * DH / 4; i += 256) {
      const int r  = i >> 5;
      const int c4 = (i & 31) * 4;
      const float4 ov = *(const float4*)(sq + r * QSTRIDE + c4);
      const float4 w4 = *(const float4*)(normw + c4);
      const float sc = snr[r];
      float4 res;
      res.x = ov.x * sc * w4.x;
      res.y = ov.y * sc * w4.y;
      res.z = ov.z * sc * w4.z;
      res.w = ov.w * sc * w4.w;
      *(float4*)(O + base + (size_t)(s0 + r) * HIDDEN + c4) = res;
    }
    __syncthreads();  // protect sq/sk/sv before next chunk restages
  }
}

// ---------------------------------------------------------------------------
extern "C" void kernel_launch(void* const* d_in, const int* in_sizes, int n_in,
                              void* d_out, int out_size, void* d_ws, size_t ws_size,
                              hipStream_t stream) {
  const float* hidden = (const float*)d_in[0];
  const float* Wq     = (const float*)d_in[1];
  const float* Wk     = (const float*)d_in[2];
  const float* Wv     = (const float*)d_in[3];
  const float* Wo     = (const float*)d_in[4];
  const float* normw  = (const float*)d_in[5];
  float* out = (float*)d_out;

  const size_t tensorElems = (size_t)ROWS * HIDDEN;  // 16M floats = 64MB
  float* q = (float*)d_ws;
  float* k = q + tensorElems;
  float* v = k + tensorElems;
  float* o = v + tensorElems;

  const size_t smemBytes = (size_t)SMEM_FLOATS * sizeof(float);
  static_assert(SMEM_FLOATS * sizeof(float) < 320 * 1024, "fits WGP LDS");
  (void)hipFuncSetAttribute((const void*)linattn_kernel,
                            hipFuncAttributeMaxDynamicSharedMemorySize,
                            (int)smemBytes);

  dim3 gg(ROWS / 64, HIDDEN / 64);  // (256, 16)
  dim3 gb(128);
  gemm_xwt_kernel<<<gg, gb, 0, stream>>>(hidden, Wq, q);
  gemm_xwt_kernel<<<gg, gb, 0, stream>>>(hidden, Wk, k);
  gemm_xwt_kernel<<<gg, gb, 0, stream>>>(hidden, Wv, v);

  linattn_kernel<<<dim3(NB * NH), dim3(256), smemBytes, stream>>>(q, k, v, normw, o);

  gemm_xwt_kernel<<<gg, gb, 0, stream>>>(o, Wo, out);
}